// LightPowerLIRALayer_49555332662205
// MI455X (gfx1250) — compile-verified
//
#include <hip/hip_runtime.h>
#include <stdint.h>

#define N_ITEMS 20000
#define N_USERS 2048
#define TILE 32
#define NNZ_PER_BLOCK 8
#define PIPE 4

typedef uint32_t v4u __attribute__((ext_vector_type(4)));
typedef int      v8i __attribute__((ext_vector_type(8)));
typedef int      v4i __attribute__((ext_vector_type(4)));

#define WAITA(imm) asm volatile("s_wait_asynccnt " #imm ::: "memory")

// ---------------------------------------------------------------------------
// TDM: DMA one 32x32 f32 tile (row stride d0_stride elements) from global into
// LDS at lds_byte_addr, padded +1 DWORD per 32-DWORD row (-> [32][33] layout).
// Descriptor packing per CDNA5 ISA section 8.3-8.5. Issued once per wave.
// ---------------------------------------------------------------------------
__device__ __forceinline__ void tdm_load_tile_32x32(
    const void* gptr, uint32_t lds_byte_addr, uint32_t tensor_d0,
    uint32_t tensor_d1, uint32_t d0_stride) {
  const uint64_t ga = (uint64_t)(uintptr_t)gptr;
  v4u g0;
  g0.x = 1u;                                   // count=1 (valid user D#)
  g0.y = lds_byte_addr;                        // lds_addr
  g0.z = (uint32_t)ga;                         // global_addr[31:0]
  g0.w = (uint32_t)(ga >> 32) | 0x80000000u;   // global_addr[56:32] | type=2
  v8i g1;
  g1[0] = 0x01120000;                          // data_size=4B | pad_enable |
                                               // pad_interval=4 (32 DWORDs),
                                               // pad_amount=0 (+1 DWORD)
  g1[1] = (int)(tensor_d0 << 16);              // tensor_dim0[15:0]
  g1[2] = (int)((tensor_d0 >> 16) | (tensor_d1 << 16));
  g1[3] = (int)((tensor_d1 >> 16) | (32u << 16));  // tile_dim0 = 32
  g1[4] = 32;                                  // tile_dim1 = 32, tile_dim2 = 0
  g1[5] = (int)d0_stride;                      // tensor_dim0_stride[31:0]
  g1[6] = 0;                                   // stride hi / dim1_stride lo
  g1[7] = 0;
  v4i g2 = {0, 0, 0, 0};
  v4i g3 = {0, 0, 0, 0};
  asm volatile("tensor_load_to_lds %0, %1, %2, %3"
               :: "s"(g0), "s"(g1), "s"(g2), "s"(g3)
               : "memory");
#if __has_builtin(__builtin_amdgcn_s_wait_tensorcnt)
  __builtin_amdgcn_s_wait_tensorcnt(0);
#else
  asm volatile("s_wait_tensorcnt 0x0" ::: "memory");
#endif
}

// ---------------------------------------------------------------------------
// X[u0+u][i] -> XT[i][u_local].  TDM pulls the tile (user-major, padded) into
// LDS; lanes write back transposed via async LDS->global stores.
// grid = (N_ITEMS/32, uc/32), block = (32, 8)
// ---------------------------------------------------------------------------
__global__ __launch_bounds__(256) void k_transpose_in(
    const float* __restrict__ X, float* __restrict__ XT, int u0, int uc) {
  __shared__ float t[TILE][TILE + 1];          // t[user][item]
  const int i0 = blockIdx.x * TILE;
  const int ub = blockIdx.y * TILE;
  const int tx = threadIdx.x, ty = threadIdx.y;

  if (ty == 0) {
    tdm_load_tile_32x32(X + (size_t)(u0 + ub) * N_ITEMS + i0,
                        (uint32_t)(uintptr_t)&t[0][0],
                        N_ITEMS, N_USERS, N_ITEMS);
  }
  __syncthreads();

  const unsigned long long xt_base = (unsigned long long)(uintptr_t)XT;
#pragma unroll
  for (int k = 0; k < 4; ++k) {
    const int il = ty + k * 8;                 // item (local)
    const uint32_t goff =
        ((uint32_t)(i0 + il) * (uint32_t)uc + (uint32_t)(ub + tx)) * 4u;
    const uint32_t ldsa = (uint32_t)(uintptr_t)&t[tx][il];
    asm volatile("global_store_async_from_lds_b32 %0, %1, %2"
                 :: "v"(goff), "v"(ldsa), "s"(xt_base)
                 : "memory");
  }
  // S_ENDPGM performs an implicit wait-idle, draining ASYNCcnt.
}

// ---------------------------------------------------------------------------
// Z[i][u_local] -> out[u0+u][i].  TDM pulls the Z tile (item-major, padded),
// lanes store transposed rows of d_out via async LDS->global stores.
// ---------------------------------------------------------------------------
__global__ __launch_bounds__(256) void k_transpose_out(
    const float* __restrict__ Z, float* __restrict__ out, int u0, int uc) {
  __shared__ float t[TILE][TILE + 1];          // t[item][user]
  const int i0 = blockIdx.x * TILE;
  const int ub = blockIdx.y * TILE;
  const int tx = threadIdx.x, ty = threadIdx.y;

  if (ty == 0) {
    tdm_load_tile_32x32(Z + (size_t)i0 * uc + ub,
                        (uint32_t)(uintptr_t)&t[0][0],
                        (uint32_t)uc, N_ITEMS, (uint32_t)uc);
  }
  __syncthreads();

  const unsigned long long out_base = (unsigned long long)(uintptr_t)out;
#pragma unroll
  for (int k = 0; k < 4; ++k) {
    const int ul = ty + k * 8;                 // user (local)
    const uint32_t goff =
        ((uint32_t)(u0 + ub + ul) * (uint32_t)N_ITEMS + (uint32_t)(i0 + tx)) * 4u;
    const uint32_t ldsa = (uint32_t)(uintptr_t)&t[tx][ul];
    asm volatile("global_store_async_from_lds_b32 %0, %1, %2"
                 :: "v"(goff), "v"(ldsa), "s"(out_base)
                 : "memory");
  }
}

// ---------------------------------------------------------------------------
// COO scatter in the transposed domain:
//   for each nnz (r,c,v):  Z[r, 0..uc) += v * XT[c, 0..uc)
// One block per 8 nnz.  XT rows are staged through a 4-deep LDS ring with
// async global->LDS b128 copies (up to 3 in flight, s_wait_asynccnt ordering;
// each wave touches only LDS its own lanes wrote -> no barriers).
// Accumulation: hardware global_atomic_add_f32, coalesced along users.
// blockDim.x = uc/4 (one float4 per lane).
// ---------------------------------------------------------------------------
__global__ __launch_bounds__(512) void k_scatter(
    const float* __restrict__ vals, const int* __restrict__ rows,
    const int* __restrict__ cols, const float* __restrict__ XT,
    float* __restrict__ Z, int uc, int nnz) {
  __shared__ float buf[PIPE][N_USERS];         // 32 KB ring
  const int tid = threadIdx.x;
  const int base = blockIdx.x * NNZ_PER_BLOCK;
  if (base >= nnz) return;

  const unsigned long long sbase = (unsigned long long)(uintptr_t)XT;
  const uint32_t lane_bytes = 16u * (uint32_t)tid;   // this lane's 4 users
  const uint32_t lds_base = (uint32_t)(uintptr_t)&buf[0][0] + lane_bytes;

  auto issue = [&](int c, int slot) {
    const uint32_t goff = (uint32_t)c * (uint32_t)uc * 4u + lane_bytes;
    const uint32_t laddr = lds_base + (uint32_t)slot * (N_USERS * 4u);
    asm volatile("global_load_async_to_lds_b128 %0, %1, %2"
                 :: "v"(laddr), "v"(goff), "s"(sbase)
                 : "memory");
  };
  auto consume = [&](int slot, int r, float v) {
    const float4 x =
        *reinterpret_cast<const float4*>(&buf[slot][4 * tid]);
    float* zp = Z + (size_t)r * uc + 4 * tid;
    unsafeAtomicAdd(zp + 0, v * x.x);
    unsafeAtomicAdd(zp + 1, v * x.y);
    unsafeAtomicAdd(zp + 2, v * x.z);
    unsafeAtomicAdd(zp + 3, v * x.w);
  };

  if (base + NNZ_PER_BLOCK <= nnz) {
    // Fast path: full block, all slots/waits compile-time constant.
    int   rj[NNZ_PER_BLOCK];
    int   cj[NNZ_PER_BLOCK];
    float vj[NNZ_PER_BLOCK];
#pragma unroll
    for (int j = 0; j < NNZ_PER_BLOCK; ++j) {
      rj[j] = rows[base + j];
      cj[j] = cols[base + j];
      vj[j] = vals[base + j];
    }
#pragma unroll
    for (int j = 0; j < PIPE - 1; ++j) issue(cj[j], j);
#pragma unroll
    for (int j = 0; j < NNZ_PER_BLOCK; ++j) {
      if (j + PIPE - 1 < NNZ_PER_BLOCK)
        issue(cj[j + PIPE - 1], (j + PIPE - 1) & (PIPE - 1));
      const int rem = NNZ_PER_BLOCK - 1 - j;   // copies allowed in flight
      if (rem >= 3)      WAITA(0x3);
      else if (rem == 2) WAITA(0x2);
      else if (rem == 1) WAITA(0x1);
      else               WAITA(0x0);
      consume(j & (PIPE - 1), rj[j], vj[j]);
    }
  } else {
    // Tail path (nnz % NNZ_PER_BLOCK != 0): simple issue/wait/consume.
    const int n = nnz - base;
    for (int j = 0; j < n; ++j) {
      const int r = rows[base + j];
      const int c = cols[base + j];
      const float v = vals[base + j];
      issue(c, j & (PIPE - 1));
      WAITA(0x0);
      consume(j & (PIPE - 1), r, v);
    }
  }
}

// ---------------------------------------------------------------------------
// Fallback if workspace is too small for the transposed pipeline.
// ---------------------------------------------------------------------------
__global__ __launch_bounds__(256) void k_scatter_direct(
    const float* __restrict__ vals, const int* __restrict__ rows,
    const int* __restrict__ cols, const float* __restrict__ X,
    float* __restrict__ out, int nnz) {
  const int k = blockIdx.x;
  if (k >= nnz) return;
  const int r = rows[k];
  const int c = cols[k];
  const float v = vals[k];
  for (int u = threadIdx.x; u < N_USERS; u += blockDim.x) {
    unsafeAtomicAdd(&out[(size_t)u * N_ITEMS + r],
                    v * X[(size_t)u * N_ITEMS + c]);
  }
}

// ---------------------------------------------------------------------------
extern "C" void kernel_launch(void* const* d_in, const int* in_sizes, int n_in,
                              void* d_out, int out_size, void* d_ws,
                              size_t ws_size, hipStream_t stream) {
  const float* X    = (const float*)d_in[0];
  const float* vals = (const float*)d_in[1];
  const int*   rows = (const int*)d_in[2];
  const int*   cols = (const int*)d_in[3];
  float*       out  = (float*)d_out;
  const int    nnz  = in_sizes[1];

  // Pick the largest user-chunk (power of two, >=256) whose XT+Z fit in ws.
  int uc = 0;
  for (int c = N_USERS; c >= 256; c >>= 1) {
    const size_t need = (size_t)2 * N_ITEMS * (size_t)c * sizeof(float);
    if (need <= ws_size) { uc = c; break; }
  }

  if (uc == 0) {
    hipMemsetAsync(d_out, 0, (size_t)out_size * sizeof(float), stream);
    k_scatter_direct<<<nnz, 256, 0, stream>>>(vals, rows, cols, X, out, nnz);
    return;
  }

  float* XT = (float*)d_ws;                       // [N_ITEMS][uc]
  float* Z  = XT + (size_t)N_ITEMS * (size_t)uc;  // [N_ITEMS][uc]

  const int  passes = N_USERS / uc;
  const dim3 tb(TILE, 8);
  const dim3 tg(N_ITEMS / TILE, uc / TILE);       // 20000 % 32 == 0
  const int  sblk  = uc / 4;                      // one float4 per lane
  const int  sgrid = (nnz + NNZ_PER_BLOCK - 1) / NNZ_PER_BLOCK;

  for (int p = 0; p < passes; ++p) {
    const int u0 = p * uc;
    k_transpose_in<<<tg, tb, 0, stream>>>(X, XT, u0, uc);
    hipMemsetAsync(Z, 0, (size_t)N_ITEMS * (size_t)uc * sizeof(float), stream);
    k_scatter<<<sgrid, sblk, 0, stream>>>(vals, rows, cols, XT, Z, uc, nnz);
    k_transpose_out<<<tg, tb, 0, stream>>>(Z, out, u0, uc);
  }
}